// SGAT_32392643346783
// MI455X (gfx1250) — compile-verified
//
#include <hip/hip_runtime.h>
#include <hip/hip_bf16.h>

// ---------------------------------------------------------------------------
// SGConv(K=2) + SELU + 2-head GAT pipeline for MI455X (gfx1250, wave32).
//  - GEMMs use V_WMMA_F32_16X16X4_F32 (fp32-exact matrix pipe).
//  - Linear op hoisted before propagation: (A^2 x)W == A^2 (xW)  -> 8x less
//    scatter traffic (64-ch instead of 512-ch node features).
//  - Node feature tables (12.8MB / 25.6MB) are L2-resident (192MB L2).
// ---------------------------------------------------------------------------

typedef __attribute__((ext_vector_type(2))) float v2f;
typedef __attribute__((ext_vector_type(8))) float v8f;

#define CDIV(a, b) (((a) + (b) - 1) / (b))

static constexpr float SELU_SCALE = 1.0507009873554805f;
static constexpr float SELU_ALPHA = 1.6732632423543772f;
static constexpr float NEG_SLOPE  = 0.2f;

// ---------------------------------------------------------------- utilities
__global__ void k_fill(float* __restrict__ p, float v, size_t n) {
  size_t i = (size_t)blockIdx.x * blockDim.x + threadIdx.x;
  if (i < n) p[i] = v;
}

__device__ __forceinline__ void atomicMaxF(float* addr, float v) {
  // IEEE-754 order-preserving integer trick (works with mixed signs).
  if (v >= 0.0f) atomicMax((int*)addr, __float_as_int(v));
  else           atomicMin((unsigned int*)addr, __float_as_uint(v));
}

// ------------------------------------------------------- degree / gcn_norm
__global__ void k_deg_accum(const int* __restrict__ ei, const float* __restrict__ ea,
                            float* __restrict__ deg, int E) {
  int e = blockIdx.x * blockDim.x + threadIdx.x;
  if (e < E) atomicAdd(&deg[ei[E + e]], ea[e]);   // dst = ei[E + e]
}

__global__ void k_dinv(float* __restrict__ dinv, int N) {
  int i = blockIdx.x * blockDim.x + threadIdx.x;
  if (i < N) { float d = dinv[i]; dinv[i] = d > 0.0f ? rsqrtf(d) : 0.0f; }
}

__global__ void k_coef(const int* __restrict__ ei, const float* __restrict__ ea,
                       const float* __restrict__ dinv, float* __restrict__ coef, int E) {
  int e = blockIdx.x * blockDim.x + threadIdx.x;
  if (e < E) coef[e] = dinv[ei[e]] * ea[e] * dinv[ei[E + e]];
}

// ------------------------------------------------------------- WMMA GEMM
// C[M,Nf] = A[M,K] @ B[K,Nf], fp32, one wave per 16x16 C tile.
// Fragment mapping (ISA 7.12.2, 32-bit):  half = lane>>4, idx = lane&15
//   A[v]  = A[m0+idx][k0 + v + 2*half]
//   B[v]  = B[k0 + v + 2*half][n0+idx]
//   C[i] -> C[m0 + i + 8*half][n0+idx]
__global__ void k_wmma_gemm(const float* __restrict__ A, const float* __restrict__ B,
                            float* __restrict__ C, int Mt /*M/16*/, int K, int Nf) {
  int wave = blockIdx.x * (blockDim.x >> 5) + (threadIdx.x >> 5);
  int ntn  = Nf >> 4;
  if (wave >= Mt * ntn) return;                // wave-uniform: EXEC stays all-1s
  int tm = wave / ntn, tn = wave - tm * ntn;
  int lane = threadIdx.x & 31;
  int half = lane >> 4;
  int idx  = lane & 15;
  size_t m0 = (size_t)tm << 4;
  int n0 = tn << 4;

  const float* arow = A + (m0 + idx) * (size_t)K;  // K even -> 8B aligned pairs
  const float* bcol = B + n0 + idx;
  v8f acc = {};
#pragma unroll 4
  for (int k0 = 0; k0 < K; k0 += 4) {
    int ka = k0 + (half << 1);
    float2 af = *(const float2*)(arow + ka);
    v2f a; a.x = af.x; a.y = af.y;
    v2f b; b.x = bcol[(size_t)ka * Nf];
           b.y = bcol[(size_t)(ka + 1) * Nf];
    acc = __builtin_amdgcn_wmma_f32_16x16x4_f32(
        /*neg_a=*/false, a, /*neg_b=*/false, b,
        /*c_mod=*/(short)0, acc, /*reuse_a=*/false, /*reuse_b=*/false);
  }
  float* crow = C + (m0 + (size_t)(half << 3)) * Nf + n0 + idx;
#pragma unroll
  for (int i = 0; i < 8; ++i) crow[(size_t)i * Nf] = acc[i];
}

// ----------------------------------------------------- SGConv propagation
// tout[n] = dinv[n]^2 * tin[n]   (self-loop term of gcn_norm)
__global__ void k_hop_self(const float* __restrict__ tin, const float* __restrict__ dinv,
                           float* __restrict__ tout, int N) {
  size_t i = (size_t)blockIdx.x * blockDim.x + threadIdx.x;
  if (i < (size_t)N * 64) {
    float di = dinv[i >> 6];
    tout[i] = di * di * tin[i];
  }
}

// tout[dst] += coef[e] * tin[src], 32 lanes/edge, float2 per lane
__global__ void k_hop_scatter(const int* __restrict__ ei, const float* __restrict__ coef,
                              const float* __restrict__ tin, float* __restrict__ tout, int E) {
  size_t t = (size_t)blockIdx.x * blockDim.x + threadIdx.x;
  if (t >= (size_t)E * 32) return;
  int e = (int)(t >> 5), j = (int)(t & 31);
  int s = ei[e], d = ei[E + e];
  float c = coef[e];
  float2 v = *(const float2*)(tin + (size_t)s * 64 + j * 2);
  float* o = tout + (size_t)d * 64 + j * 2;
  atomicAdd(o + 0, c * v.x);
  atomicAdd(o + 1, c * v.y);
}

__global__ void k_bias_selu(float* __restrict__ t, const float* __restrict__ b, size_t total) {
  size_t i = (size_t)blockIdx.x * blockDim.x + threadIdx.x;
  if (i < total) {
    float x = t[i] + b[i & 63];
    t[i] = SELU_SCALE * (x > 0.0f ? x : SELU_ALPHA * (expf(x) - 1.0f));
  }
}

// ----------------------------------------------------------------- GAT
// per (node, head): a_src = <g[n,h,:], att_src[h,:]>, same for a_dst
__global__ void k_att(const float* __restrict__ g, const float* __restrict__ att_s,
                      const float* __restrict__ att_d, float* __restrict__ asrc,
                      float* __restrict__ adst, int N) {
  int i = blockIdx.x * blockDim.x + threadIdx.x;
  if (i >= N * 2) return;
  int n = i >> 1, h = i & 1;
  const float* gp = g + (size_t)n * 128 + h * 64;
  const float* as = att_s + h * 64;
  const float* ad = att_d + h * 64;
  float s0 = 0.0f, s1 = 0.0f;
#pragma unroll 8
  for (int c = 0; c < 64; ++c) { float gv = gp[c]; s0 += gv * as[c]; s1 += gv * ad[c]; }
  asrc[i] = s0; adst[i] = s1;
}

// virtual edge range [0, E+N): ep < E real edge, else self-loop at node ep-E
__global__ void k_edge_max(const int* __restrict__ ei, const float* __restrict__ asrc,
                           const float* __restrict__ adst, float* __restrict__ emax,
                           int E, int N) {
  int ep = blockIdx.x * blockDim.x + threadIdx.x;
  if (ep >= E + N) return;
  int s = ep < E ? ei[ep]     : ep - E;
  int d = ep < E ? ei[E + ep] : ep - E;
#pragma unroll
  for (int h = 0; h < 2; ++h) {
    float l = asrc[s * 2 + h] + adst[d * 2 + h];
    l = l > 0.0f ? l : NEG_SLOPE * l;
    atomicMaxF(&emax[d * 2 + h], l);
  }
}

__global__ void k_edge_expsum(const int* __restrict__ ei, const float* __restrict__ asrc,
                              const float* __restrict__ adst, const float* __restrict__ emax,
                              float* __restrict__ denom, float* __restrict__ alpha,
                              int E, int N) {
  int ep = blockIdx.x * blockDim.x + threadIdx.x;
  if (ep >= E + N) return;
  int s = ep < E ? ei[ep]     : ep - E;
  int d = ep < E ? ei[E + ep] : ep - E;
#pragma unroll
  for (int h = 0; h < 2; ++h) {
    float l = asrc[s * 2 + h] + adst[d * 2 + h];
    l = l > 0.0f ? l : NEG_SLOPE * l;
    float ee = expf(l - emax[d * 2 + h]);
    alpha[(size_t)ep * 2 + h] = ee;
    atomicAdd(&denom[d * 2 + h], ee);
  }
}

// acc[dst,h,:] += (alpha/denom) * g[src,h,:] ; 32 lanes/edge, float4 per lane
__global__ void k_gat_scatter(const int* __restrict__ ei, const float* __restrict__ alpha,
                              const float* __restrict__ denom, const float* __restrict__ g,
                              float* __restrict__ acc, int E, int N) {
  size_t t = (size_t)blockIdx.x * blockDim.x + threadIdx.x;
  if (t >= (size_t)(E + N) * 32) return;
  int ep = (int)(t >> 5), j = (int)(t & 31);
  int s = ep < E ? ei[ep]     : ep - E;
  int d = ep < E ? ei[E + ep] : ep - E;
  int h = j >> 4;  // lanes 0-15 -> head 0 (feat 0..63), 16-31 -> head 1
  float w = alpha[(size_t)ep * 2 + h] / fmaxf(denom[d * 2 + h], 1e-16f);
  float4 gv = *(const float4*)(g + (size_t)s * 128 + j * 4);
  float* o = acc + (size_t)d * 128 + j * 4;
  atomicAdd(o + 0, w * gv.x);
  atomicAdd(o + 1, w * gv.y);
  atomicAdd(o + 2, w * gv.z);
  atomicAdd(o + 3, w * gv.w);
}

__global__ void k_finalize(const float* __restrict__ acc, const float* __restrict__ bg,
                           float* __restrict__ out, int N) {
  size_t i = (size_t)blockIdx.x * blockDim.x + threadIdx.x;
  if (i < (size_t)N * 64) {
    size_t n = i >> 6; int c = (int)(i & 63);
    out[i] = 0.5f * (acc[n * 128 + c] + acc[n * 128 + 64 + c]) + bg[c];
  }
}

// ---------------------------------------------------------------------------
extern "C" void kernel_launch(void* const* d_in, const int* in_sizes, int n_in,
                              void* d_out, int out_size, void* d_ws, size_t ws_size,
                              hipStream_t stream) {
  const float* x     = (const float*)d_in[0];
  const int*   ei    = (const int*)  d_in[1];   // [2,E] row-major (int32 per harness)
  const float* ea    = (const float*)d_in[2];
  const float* W_sg  = (const float*)d_in[3];   // [512,64]
  const float* b_sg  = (const float*)d_in[4];
  const float* W_gat = (const float*)d_in[5];   // [64,128]
  const float* att_s = (const float*)d_in[6];   // [2,64]
  const float* att_d = (const float*)d_in[7];
  const float* b_gat = (const float*)d_in[8];
  float* out = (float*)d_out;

  const int N  = in_sizes[0] / 512;   // 50000 (multiple of 16)
  const int E  = in_sizes[2];         // 1600000
  const int EP = E + N;

  // ---- workspace carve (≈ 98 MB) ----
  char* w = (char*)d_ws;
  auto carve = [&](size_t elems) -> float* {
    float* p = (float*)w;
    w += (elems * sizeof(float) + 255) & ~(size_t)255;
    return p;
  };
  float* dinv  = carve(N);           // degree -> rsqrt in place
  float* coef  = carve(E);
  float* t0    = carve((size_t)N * 64);
  float* t1    = carve((size_t)N * 64);
  float* g     = carve((size_t)N * 128);
  float* asrc  = carve((size_t)N * 2);
  float* adst  = carve((size_t)N * 2);
  float* emax  = carve((size_t)N * 2);
  float* denom = carve((size_t)N * 2);
  float* alpha = carve((size_t)EP * 2);
  float* acc   = carve((size_t)N * 128);

  const int B = 256;

  // ---- gcn_norm coefficients ----
  k_fill<<<CDIV((size_t)N, B), B, 0, stream>>>(dinv, 1.0f, (size_t)N);  // self-loop weight
  k_deg_accum<<<CDIV(E, B), B, 0, stream>>>(ei, ea, dinv, E);
  k_dinv<<<CDIV(N, B), B, 0, stream>>>(dinv, N);
  k_coef<<<CDIV(E, B), B, 0, stream>>>(ei, ea, dinv, coef, E);

  // ---- t0 = x @ W_sg  (linear hoisted before propagation) ----
  {
    int waves = (N / 16) * (64 / 16);
    k_wmma_gemm<<<CDIV(waves, B / 32), B, 0, stream>>>(x, W_sg, t0, N / 16, 512, 64);
  }

  // ---- two hops of D^-1/2 (A+I) D^-1/2 on 64-dim features ----
  k_hop_self<<<CDIV((size_t)N * 64, B), B, 0, stream>>>(t0, dinv, t1, N);
  k_hop_scatter<<<CDIV((size_t)E * 32, B), B, 0, stream>>>(ei, coef, t0, t1, E);
  k_hop_self<<<CDIV((size_t)N * 64, B), B, 0, stream>>>(t1, dinv, t0, N);
  k_hop_scatter<<<CDIV((size_t)E * 32, B), B, 0, stream>>>(ei, coef, t1, t0, E);

  // ---- bias + SELU ----
  k_bias_selu<<<CDIV((size_t)N * 64, B), B, 0, stream>>>(t0, b_sg, (size_t)N * 64);

  // ---- g = h @ W_gat ----
  {
    int waves = (N / 16) * (128 / 16);
    k_wmma_gemm<<<CDIV(waves, B / 32), B, 0, stream>>>(t0, W_gat, g, N / 16, 64, 128);
  }

  // ---- attention logits + softmax over incoming edges ----
  k_att<<<CDIV(N * 2, B), B, 0, stream>>>(g, att_s, att_d, asrc, adst, N);
  k_fill<<<CDIV((size_t)N * 2, B), B, 0, stream>>>(emax, -3.0e38f, (size_t)N * 2);
  k_fill<<<CDIV((size_t)N * 2, B), B, 0, stream>>>(denom, 0.0f, (size_t)N * 2);
  k_fill<<<CDIV((size_t)N * 128, B), B, 0, stream>>>(acc, 0.0f, (size_t)N * 128);
  k_edge_max<<<CDIV(EP, B), B, 0, stream>>>(ei, asrc, adst, emax, E, N);
  k_edge_expsum<<<CDIV(EP, B), B, 0, stream>>>(ei, asrc, adst, emax, denom, alpha, E, N);

  // ---- weighted aggregation + head mean + bias ----
  k_gat_scatter<<<CDIV((size_t)EP * 32, B), B, 0, stream>>>(ei, alpha, denom, g, acc, E, N);
  k_finalize<<<CDIV((size_t)N * 64, B), B, 0, stream>>>(acc, b_gat, out, N);
}